// SimpleGraphConvolution_23965917512253
// MI455X (gfx1250) — compile-verified
//
#include <hip/hip_runtime.h>
#include <hip/hip_bf16.h>

#define GC_N_NODES 50000
#define GC_IN_F    256
#define GC_OUT_F   128
#define GC_LDS_PITCH (GC_IN_F + 4)   // 260 floats: 4-float pad kills LDS bank conflicts

typedef __attribute__((ext_vector_type(2))) float v2f;
typedef __attribute__((ext_vector_type(8))) float v8f;

// ---------------------------------------------------------------------------
// Zero the output buffer (segment_sum identity). float4 stores, bandwidth-bound.
// ---------------------------------------------------------------------------
__global__ void gc_zero_kernel(float4* __restrict__ out, int n4) {
    int i = blockIdx.x * blockDim.x + threadIdx.x;
    if (i < n4) out[i] = make_float4(0.f, 0.f, 0.f, 0.f);
}

// ---------------------------------------------------------------------------
// support = x @ W^T + b   via V_WMMA_F32_16X16X4_F32 (full fp32 precision;
// GEMM is HBM-bound at 23.3 TB/s so fp32 WMMA is already free).
//
// Block = 256 threads = 8 waves. Block owns 16 rows of x; wave w owns output
// columns [16w, 16w+16). x tile (16x256) staged in LDS (padded rows), W rows
// read straight from global (128 KB total, L2-resident across all blocks).
//
// WMMA f32 16x16x4 fragment layout (wave32):
//   A (16x4): lane&15 = M; lanes<16 hold K={k,k+1} in v0/v1, lanes>=16 K={k+2,k+3}
//   B (4x16): lane&15 = N; same K split across lane halves
//   C/D (16x16): VGPR r, lane l -> element (M = r + 8*(l>>4), N = l&15)
// ---------------------------------------------------------------------------
__launch_bounds__(256)
__global__ void gc_gemm_kernel(const float* __restrict__ x,
                               const float* __restrict__ W,
                               const float* __restrict__ b,
                               float* __restrict__ support) {
    __shared__ float xs[16 * GC_LDS_PITCH];

    const int tid  = threadIdx.x;
    const int row0 = blockIdx.x * 16;

    // Cooperative coalesced load of the 16x256 x tile into padded LDS.
    // 256 threads x 4 float4 = 4096 floats. LDS row stride 260 (16B aligned).
    {
        const int r4 = tid >> 6;          // 0..3
        const int c4 = (tid & 63) * 4;    // 0..252
        #pragma unroll
        for (int i = 0; i < 4; ++i) {
            const int row = r4 + 4 * i;
            *(float4*)(&xs[row * GC_LDS_PITCH + c4]) =
                *(const float4*)(x + (size_t)(row0 + row) * GC_IN_F + c4);
        }
    }
    __syncthreads();

    const int wave  = tid >> 5;          // 0..7 -> column strip
    const int lane  = tid & 31;
    const int hlane = lane & 15;         // M (for A) / N (for B,C,D)
    const int khalf = lane >> 4;         // 0: K={k,k+1}, 1: K={k+2,k+3}
    const int ncol0 = wave * 16;

    const float* xrow = xs + hlane * GC_LDS_PITCH + 2 * khalf;            // LDS A source
    const float* wrow = W + (size_t)(ncol0 + hlane) * GC_IN_F + 2 * khalf; // global B source

    v8f acc = {};
    #pragma unroll 4
    for (int k = 0; k < GC_IN_F; k += 4) {
        v2f afrag = *(const v2f*)(xrow + k);   // ds_load_b64, conflict-free
        v2f bfrag = *(const v2f*)(wrow + k);   // global_load_b64, L2-resident
        acc = __builtin_amdgcn_wmma_f32_16x16x4_f32(
                  false, afrag, false, bfrag, (short)0, acc, false, false);
    }

    // Bias: every element in this lane has N = ncol0 + hlane.
    const float bias = b[ncol0 + hlane];
    #pragma unroll
    for (int r = 0; r < 8; ++r) acc[r] += bias;

    // Store D: element (M = r + 8*khalf, N = hlane) per VGPR r.
    float* outp = support + (size_t)(row0 + 8 * khalf) * GC_OUT_F + ncol0 + hlane;
    #pragma unroll
    for (int r = 0; r < 8; ++r) outp[(size_t)r * GC_OUT_F] = acc[r];
}

// ---------------------------------------------------------------------------
// Edge scatter: out[row] += val * support[col].  One wave per edge; each lane
// handles 4 contiguous features (float4 gather + 4 f32 atomics). support and
// out both fit in the 192 MB L2, so gathers and atomics run at L2 speed.
// ---------------------------------------------------------------------------
__launch_bounds__(256)
__global__ void gc_scatter_kernel(const float* __restrict__ support,
                                  const float* __restrict__ edge_val,
                                  const int* __restrict__ edge_row,
                                  const int* __restrict__ edge_col,
                                  float* __restrict__ out,
                                  int n_edges) {
    const int lane = threadIdx.x & 31;
    const int e = blockIdx.x * 8 + (threadIdx.x >> 5);
    if (e >= n_edges) return;

    const int   r = edge_row[e];
    const int   c = edge_col[e];
    const float v = edge_val[e];

    const float4 s = *(const float4*)(support + (size_t)c * GC_OUT_F + lane * 4);
    float* o = out + (size_t)r * GC_OUT_F + lane * 4;
    atomicAdd(o + 0, v * s.x);
    atomicAdd(o + 1, v * s.y);
    atomicAdd(o + 2, v * s.z);
    atomicAdd(o + 3, v * s.w);
}

// ---------------------------------------------------------------------------
extern "C" void kernel_launch(void* const* d_in, const int* in_sizes, int n_in,
                              void* d_out, int out_size, void* d_ws, size_t ws_size,
                              hipStream_t stream) {
    const float* x        = (const float*)d_in[0];  // [N_NODES, IN_F]
    const float* W        = (const float*)d_in[1];  // [OUT_F, IN_F]
    const float* b        = (const float*)d_in[2];  // [OUT_F]
    const float* edge_val = (const float*)d_in[3];  // [E]
    const int*   edge_row = (const int*)d_in[4];    // [E]
    const int*   edge_col = (const int*)d_in[5];    // [E]
    float*       out      = (float*)d_out;          // [N_NODES, OUT_F]
    float*       support  = (float*)d_ws;           // [N_NODES, OUT_F] scratch (25.6 MB)
    const int    n_edges  = in_sizes[3];

    // 1) out = 0
    const int n4 = out_size / 4;
    gc_zero_kernel<<<(n4 + 255) / 256, 256, 0, stream>>>((float4*)out, n4);

    // 2) support = x @ W^T + b   (50000 rows / 16 per block = 3125 blocks, exact)
    gc_gemm_kernel<<<GC_N_NODES / 16, 256, 0, stream>>>(x, W, b, support);

    // 3) atomic edge scatter (8 edges per 256-thread block)
    gc_scatter_kernel<<<(n_edges + 7) / 8, 256, 0, stream>>>(
        support, edge_val, edge_row, edge_col, out, n_edges);
}